// FactoredBlock_17454747091330
// MI455X (gfx1250) — compile-verified
//
#include <hip/hip_runtime.h>

typedef __attribute__((ext_vector_type(2))) float v2f;
typedef __attribute__((ext_vector_type(8))) float v8f;

#define INTER      768
#define OUTC       256
#define TILE_M     16
#define LDS_STRIDE (INTER + 4)   // 772: row stride ≡ 4 (mod 64 banks) -> conflict-free column reads
#define BLOCK      256

__global__ __launch_bounds__(BLOCK) void factored_block_kernel(
    const int*   __restrict__ batch_idx,   // [nnz] sorted
    const int*   __restrict__ active_idx,  // [nnz]
    const float* __restrict__ values,      // [nnz]
    const int*   __restrict__ f_table,     // [49152] = i % 768
    const float* __restrict__ W,           // [768, 256]
    float*       __restrict__ out,         // [N, 256]
    int nnz, int N)
{
    __shared__ float xs[TILE_M * LDS_STRIDE];   // 49,408 B of the 320 KB WGP LDS
    __shared__ int   seg[2];                    // [lo, hi) broadcast

    const int tid  = threadIdx.x;
    const int row0 = blockIdx.x * TILE_M;

    // ---- 1) zero the dense LDS tile -------------------------------------
    for (int i = tid; i < TILE_M * LDS_STRIDE; i += BLOCK) xs[i] = 0.0f;

    // ---- 2) one thread binary-searches sorted batch_idx for the tile ----
    if (tid == 0) {
        int l = 0, r = nnz;
        while (l < r) { int m = (l + r) >> 1; if (batch_idx[m] < row0) l = m + 1; else r = m; }
        seg[0] = l;
        r = nnz;
        while (l < r) { int m = (l + r) >> 1; if (batch_idx[m] < row0 + TILE_M) l = m + 1; else r = m; }
        seg[1] = l;
    }
    __syncthreads();

    // ---- 3) scatter-sum nonzeros into LDS (duplicates -> ds_add_f32) ----
    const int lo = seg[0];
    const int hi = seg[1];
    for (int k = lo + tid; k < hi; k += BLOCK) {
        int r = batch_idx[k] - row0;
        int c = f_table[active_idx[k]];
        atomicAdd(&xs[r * LDS_STRIDE + c], values[k]);
    }
    __syncthreads();

    // ---- 4) WMMA: each wave owns two 16-wide column tiles ---------------
    const int wave = tid >> 5;        // 0..7
    const int lane = tid & 31;
    const int half = lane >> 4;       // 0: lanes 0-15, 1: lanes 16-31
    const int l15  = lane & 15;
    const int n0   = wave * 32;       // this wave covers cols [n0, n0+32)

    v8f acc0 = {};
    v8f acc1 = {};

    const float* xrow = &xs[l15 * LDS_STRIDE + 2 * half];
    const float* wcol = W + 2 * half * OUTC + l15;

    for (int k0 = 0; k0 < INTER; k0 += 4) {
        // A fragment (16x4 f32): lane l15 = row M, {VGPR0,VGPR1} = {ka, ka+1}
        v2f a = *(const v2f*)&xrow[k0];

        // B fragments (4x16 f32): lane l15 = col N, {VGPR0,VGPR1} = rows {ka, ka+1}
        const float* wr = wcol + k0 * OUTC;
        v2f b0, b1;
        b0.x = wr[n0];             b0.y = wr[OUTC + n0];
        b1.x = wr[n0 + 16];        b1.y = wr[OUTC + n0 + 16];

        acc0 = __builtin_amdgcn_wmma_f32_16x16x4_f32(
                   false, a, false, b0, (short)0, acc0, false, false);
        acc1 = __builtin_amdgcn_wmma_f32_16x16x4_f32(
                   false, a, false, b1, (short)0, acc1, false, false);
    }

    // ---- 5) store D: VGPR r -> M = r + 8*half, N = l15 ------------------
    float* obase = out + (size_t)(row0 + 8 * half) * OUTC + n0 + l15;
    if (row0 + TILE_M <= N) {
        // uniform fast path: full tile in range -> straight-line stores
        #pragma unroll
        for (int r = 0; r < 8; ++r) {
            obase[r * OUTC]      = acc0[r];
            obase[r * OUTC + 16] = acc1[r];
        }
    } else {
        #pragma unroll
        for (int r = 0; r < 8; ++r) {
            if (row0 + 8 * half + r < N) {
                obase[r * OUTC]      = acc0[r];
                obase[r * OUTC + 16] = acc1[r];
            }
        }
    }
}

extern "C" void kernel_launch(void* const* d_in, const int* in_sizes, int n_in,
                              void* d_out, int out_size, void* d_ws, size_t ws_size,
                              hipStream_t stream) {
    const int*   batch_idx  = (const int*)d_in[0];
    const int*   active_idx = (const int*)d_in[1];
    const float* values     = (const float*)d_in[2];
    const int*   f_table    = (const int*)d_in[3];
    const float* W          = (const float*)d_in[4];
    float*       out        = (float*)d_out;

    const int nnz = in_sizes[0];
    const int N   = out_size / OUTC;     // avoids reading the device-side N scalar
    const int grid = (N + TILE_M - 1) / TILE_M;

    factored_block_kernel<<<grid, BLOCK, 0, stream>>>(
        batch_idx, active_idx, values, f_table, W, out, nnz, N);
}